// T5ExtractorEncoder_12386685682449
// MI455X (gfx1250) — compile-verified
//
#include <hip/hip_runtime.h>

typedef __attribute__((ext_vector_type(2))) float v2f;
typedef __attribute__((ext_vector_type(4))) float v4f;
typedef __attribute__((ext_vector_type(8))) float v8f;

#define NEG_INF_F (-1e9f)

constexpr int Bn = 8;      // batch
constexpr int Sn = 4096;   // sequence
constexpr int Dn = 1024;   // hidden dim
constexpr int Nn = 128;    // num sentences
constexpr int Kn = 8;      // extraction_k

// ---------------------------------------------------------------------------
// Kernel 1: token_dot[b,s] = hidden[b,s,:] . w  via V_WMMA_F32_16X16X4_F32
// A = 16 tokens x 4 d-values (fp32), B = w-chunk replicated over 16 columns.
// Each wave owns a 16-token tile; 8 waves/block -> 128 tokens/block.
// ---------------------------------------------------------------------------
__global__ __launch_bounds__(256) void token_dot_wmma(
    const float* __restrict__ hidden, const float* __restrict__ w,
    float* __restrict__ token_dot) {
  const int wave = threadIdx.x >> 5;     // 0..7
  const int lane = threadIdx.x & 31;
  const int half = lane >> 4;            // K-half select (ISA A/B layout)
  const int mrow = lane & 15;            // token row within tile
  const int base = blockIdx.x * 128 + wave * 16;

  const float* arow = hidden + (size_t)(base + mrow) * Dn + 2 * half;
  const float* wrow = w + 2 * half;

  v8f c = {};
#pragma unroll 4
  for (int d = 0; d < Dn; d += 4) {
    // A: lane(half h, row m): VGPR0 = A[m][2h], VGPR1 = A[m][2h+1]
    v2f a = *(const v2f*)(arow + d);
    // B rows are constant (all 16 columns identical): VGPRr = w[d + 2h + r]
    v2f b;
    b.x = wrow[d];
    b.y = wrow[d + 1];
    c = __builtin_amdgcn_wmma_f32_16x16x4_f32(
        /*neg_a=*/false, a, /*neg_b=*/false, b,
        /*c_mod=*/(short)0, c, /*reuse_a=*/false, /*reuse_b=*/false);
  }

  // C/D layout: lane(col n, half h), VGPR r = D[r + 8h][n]; all columns equal.
  if (mrow == 0) {
    const int t0 = base + 8 * half;
#pragma unroll
    for (int r = 0; r < 8; ++r) token_dot[t0 + r] = c[r];
  }
}

// ---------------------------------------------------------------------------
// Kernel 2: deterministic segment reduction.
// One block per (b, n): fixed-order strided accumulate + shared tree reduce.
// ---------------------------------------------------------------------------
__global__ __launch_bounds__(256) void seg_reduce(
    const float* __restrict__ token_dot, const int* __restrict__ ind,
    float* __restrict__ seg_dot, float* __restrict__ counts) {
  const int b = blockIdx.x / Nn;
  const int n = blockIdx.x % Nn;
  __shared__ float sdot[256];
  __shared__ float scnt[256];

  const int* indb = ind + b * Sn;
  const float* tdb = token_dot + b * Sn;
  float acc = 0.f, cnt = 0.f;
  for (int s = threadIdx.x; s < Sn; s += 256) {
    if (indb[s] == n) { acc += tdb[s]; cnt += 1.f; }
  }
  sdot[threadIdx.x] = acc;
  scnt[threadIdx.x] = cnt;
  __syncthreads();
  for (int off = 128; off > 0; off >>= 1) {
    if (threadIdx.x < off) {
      sdot[threadIdx.x] += sdot[threadIdx.x + off];
      scnt[threadIdx.x] += scnt[threadIdx.x + off];
    }
    __syncthreads();
  }
  if (threadIdx.x == 0) {
    seg_dot[blockIdx.x] = sdot[0];
    counts[blockIdx.x] = scnt[0];
  }
}

// ---------------------------------------------------------------------------
// Kernel 3: K-step greedy selection. One block per batch, one thread per n.
// Argmax tree breaks ties toward the lowest index (jnp.argmax semantics).
// ---------------------------------------------------------------------------
__global__ __launch_bounds__(128) void select_k(
    const float* __restrict__ seg_dot, const float* __restrict__ counts,
    const float* __restrict__ cls_b_ptr,
    float* __restrict__ out_selected,   // [B,N]
    float* __restrict__ out_logits) {   // [K,B,N]
  const int b = blockIdx.x;
  const int n = threadIdx.x;
  __shared__ float sval[128];
  __shared__ int sidx[128];
  __shared__ float s_curdot, s_curlen;

  const float sd = seg_dot[b * Nn + n];
  const float cv = counts[b * Nn + n];
  const float len = (cv < 1.f) ? 1.f : cv;   // clamp(min=1)
  float mask = (cv > 0.f) ? 1.f : 0.f;
  float selv = 0.f;
  const float clsb = cls_b_ptr[0];

  if (n == 0) { s_curdot = 0.f; s_curlen = 0.f; }
  __syncthreads();

  for (int k = 0; k < Kn; ++k) {
    float logit = (s_curdot + sd) / (s_curlen + len) + clsb;
    logit = (mask > 0.f) ? logit : NEG_INF_F;
    out_logits[(k * Bn + b) * Nn + n] = logit;
    sval[n] = logit;
    sidx[n] = n;
    __syncthreads();
    for (int off = 64; off > 0; off >>= 1) {
      if (n < off) {
        const float v2 = sval[n + off];
        const int i2 = sidx[n + off];
        if (v2 > sval[n] || (v2 == sval[n] && i2 < sidx[n])) {
          sval[n] = v2;
          sidx[n] = i2;
        }
      }
      __syncthreads();
    }
    const int win = sidx[0];
    if (n == win) {
      selv = 1.f;          // sel accumulate + clip(<=1)
      mask = 0.f;          // smask *= (1 - one_hot)
      s_curdot += sd;      // new_sum.w = cur_sum.w + seg_sum[win].w
      s_curlen += len;     // new_len = cur_len + len[win]
    }
    __syncthreads();
  }
  out_selected[b * Nn + n] = selv;
}

// ---------------------------------------------------------------------------
// Kernel 4: gather token mask + masked copy of hidden (float4 streams).
// One block per token; 256 threads x float4 == 1024 floats.
// ---------------------------------------------------------------------------
__global__ __launch_bounds__(256) void apply_mask(
    const float* __restrict__ hidden, const int* __restrict__ ind,
    const float* __restrict__ selected,  // [B,N]
    float* __restrict__ out_mask,        // [B,S]
    float* __restrict__ out_masked) {    // [B,S,D]
  const int tok = blockIdx.x;            // 0 .. B*S-1
  const int b = tok >> 12;               // S = 4096
  const int n = ind[tok];
  const float m = selected[b * Nn + n];
  if (threadIdx.x == 0) out_mask[tok] = m;

  const v4f* src = (const v4f*)(hidden + (size_t)tok * Dn);
  v4f* dst = (v4f*)(out_masked + (size_t)tok * Dn);
  v4f v = src[threadIdx.x];
  dst[threadIdx.x] = v * m;
}

// ---------------------------------------------------------------------------
extern "C" void kernel_launch(void* const* d_in, const int* in_sizes, int n_in,
                              void* d_out, int out_size, void* d_ws, size_t ws_size,
                              hipStream_t stream) {
  (void)in_sizes; (void)n_in; (void)out_size; (void)ws_size;

  const float* hidden = (const float*)d_in[0];   // [B,S,D] f32
  const int* ind      = (const int*)d_in[1];     // [B,S] (int32; jax x64 off)
  const float* cls_w  = (const float*)d_in[2];   // [D] f32
  const float* cls_b  = (const float*)d_in[3];   // scalar f32

  // Output layout: selected [B,N] | logits [K,B,N] | mask [B,S] | masked [B,S,D]
  float* o = (float*)d_out;
  float* o_selected = o;
  float* o_logits   = o_selected + Bn * Nn;
  float* o_mask     = o_logits + Kn * Bn * Nn;
  float* o_masked   = o_mask + Bn * Sn;

  // Workspace: token_dot [B*S] | seg_dot [B*N] | counts [B*N]
  float* ws        = (float*)d_ws;
  float* token_dot = ws;
  float* seg_dot   = token_dot + Bn * Sn;
  float* cnts      = seg_dot + Bn * Nn;

  // 1) fp32 WMMA GEMV: 32768 tokens / 128 per block = 256 blocks
  token_dot_wmma<<<(Bn * Sn) / 128, 256, 0, stream>>>(hidden, cls_w, token_dot);

  // 2) deterministic segment reduction: one block per (b,n)
  seg_reduce<<<Bn * Nn, 256, 0, stream>>>(token_dot, ind, seg_dot, cnts);

  // 3) K-step selection: one block per batch
  select_k<<<Bn, 128, 0, stream>>>(seg_dot, cnts, cls_b, o_selected, o_logits);

  // 4) mask gather + masked hidden copy: one block per token
  apply_mask<<<Bn * Sn, 256, 0, stream>>>(hidden, ind, o_selected, o_mask, o_masked);
}